// SimpleSemanticEncoder_78400333021646
// MI455X (gfx1250) — compile-verified
//
#include <hip/hip_runtime.h>
#include <hip/hip_bf16.h>
#include <stdint.h>

#define VOCAB 50257
#define FEAT  512
#define EPS   1e-12f

typedef uint32_t u32;
typedef uint64_t u64;
typedef __attribute__((ext_vector_type(4))) u32   u32x4;
typedef __attribute__((ext_vector_type(8))) u32   u32x8;
typedef __attribute__((ext_vector_type(4))) float f32x4;

// ---------------------------------------------------------------------------
// Kernel 1: TDM-staged transpose  W[FEAT][VOCAB] -> Wt[VOCAB][FEAT]
// One 64x64 f32 tile per block, pulled into LDS by the Tensor Data Mover with
// 1-dword row padding (pitch 65) so transposed LDS reads are conflict-free.
// ---------------------------------------------------------------------------
#define TDIM   64
#define TPITCH 65

__global__ __launch_bounds__(256)
void tdm_transpose_kernel(const float* __restrict__ W, float* __restrict__ Wt) {
  __shared__ float tile[TDIM * TPITCH];   // 16640 B

  const u32 v0 = blockIdx.x * TDIM;       // vocab tile origin (tensor dim0, contiguous)
  const u32 f0 = blockIdx.y * TDIM;       // feat  tile origin (tensor dim1, strided)

  // LDS byte offset of tile[] within this workgroup's LDS allocation.
  u32 lds_off = (u32)(uintptr_t)(__attribute__((address_space(3))) float*)&tile[0];

  u64 gaddr    = (u64)(uintptr_t)W + ((u64)f0 * (u64)VOCAB + (u64)v0) * 4ull;
  u32 dim0_rem = (u32)VOCAB - v0;         // remaining extent -> TDM zero-fills OOB

  // ---- D# group 0 (128b): count | lds_addr | global_addr | type=2 ----
  u32x4 g0;
  g0.x = 1u;                                   // count=1, user mode, no gather
  g0.y = lds_off;                              // lds_addr [63:32]
  g0.z = (u32)(gaddr & 0xFFFFFFFFull);         // global_addr [95:64]
  g0.w = (u32)(gaddr >> 32) | (2u << 30);      // global_addr [120:96] | type=2

  // ---- D# group 1 (256b): sizes/strides/padding ----
  u32x8 g1;
  g1.s0 = (2u << 16)        // data_size = 4 bytes
        | (1u << 20)        // pad_enable
        | (5u << 22)        // pad_interval: 64 dwords (one tile row)
        | (0u << 25);       // pad_amount: 1 dword
  g1.s1 = (dim0_rem & 0xFFFFu) << 16;               // tensor_dim0[15:0]
  g1.s2 = (dim0_rem >> 16) | ((u32)FEAT << 16);     // tensor_dim0[31:16] | tensor_dim1[15:0]
  g1.s3 = ((u32)TDIM << 16);                        // tensor_dim1[31:16]=0 | tile_dim0=64
  g1.s4 = (u32)TDIM;                                // tile_dim1=64 | tile_dim2=0
  g1.s5 = (u32)VOCAB;                               // tensor_dim0_stride[31:0] (row stride of W)
  g1.s6 = 0u;                                       // stride0[47:32]=0 | dim1_stride lo=0
  g1.s7 = 0u;

  if (threadIdx.x == 0) {
    asm volatile("tensor_load_to_lds %0, %1"
                 :
                 : "s"(g0), "s"(g1)
                 : "memory");
    __builtin_amdgcn_s_wait_tensorcnt(0);
  }
  __syncthreads();

  // Write out transposed: Wt[v][f] gets the LDS tile value at row f - f0, column v - v0; coalesced along f,
  // LDS column reads stride 65 dwords -> all 64 banks distinct.
  const u32 fx  = threadIdx.x & 63;
  const u32 vy0 = threadIdx.x >> 6;
#pragma unroll
  for (u32 i = 0; i < TDIM / 4; ++i) {
    u32 vy = vy0 + i * 4;
    u32 v  = v0 + vy;
    if (v < (u32)VOCAB) {
      Wt[(u64)v * FEAT + f0 + fx] = tile[fx * TPITCH + vy];
    }
  }
}

// ---------------------------------------------------------------------------
// Kernel 2: gather + bias + L2-normalize. One wave32 per token: 4x b128 loads
// per lane cover the 2KB row, shfl_xor wave reduction, NT stores for output.
// ---------------------------------------------------------------------------
__global__ __launch_bounds__(256)
void gather_norm_kernel(const int* __restrict__ ids,
                        const float* __restrict__ Wt,
                        const float* __restrict__ bias,
                        float* __restrict__ out, int n_tokens) {
  const int lane   = threadIdx.x & 31;
  const int wave   = (int)((blockIdx.x * blockDim.x + threadIdx.x) >> 5);
  const int nwaves = (int)((gridDim.x * blockDim.x) >> 5);

  const f32x4* b4 = (const f32x4*)bias;
  f32x4 bb0 = b4[lane], bb1 = b4[lane + 32], bb2 = b4[lane + 64], bb3 = b4[lane + 96];

  for (int t = wave; t < n_tokens; t += nwaves) {
    int id = ids[t];

    // Prefetch the next token's 2KB row: 64B per lane x 32 lanes.
    int tn = t + nwaves;
    if (tn < n_tokens) {
      int idn = ids[tn];
      __builtin_prefetch((const char*)(Wt + (u64)idn * FEAT) + lane * 64, 0, 3);
    }

    const f32x4* row = (const f32x4*)(Wt + (u64)id * FEAT);
    f32x4 a = row[lane];
    f32x4 b = row[lane + 32];
    f32x4 c = row[lane + 64];
    f32x4 d = row[lane + 96];

    a.x += bb0.x; a.y += bb0.y; a.z += bb0.z; a.w += bb0.w;
    b.x += bb1.x; b.y += bb1.y; b.z += bb1.z; b.w += bb1.w;
    c.x += bb2.x; c.y += bb2.y; c.z += bb2.z; c.w += bb2.w;
    d.x += bb3.x; d.y += bb3.y; d.z += bb3.z; d.w += bb3.w;

    float ss = a.x*a.x + a.y*a.y + a.z*a.z + a.w*a.w
             + b.x*b.x + b.y*b.y + b.z*b.z + b.w*b.w
             + c.x*c.x + c.y*c.y + c.z*c.z + c.w*c.w
             + d.x*d.x + d.y*d.y + d.z*d.z + d.w*d.w;

#pragma unroll
    for (int m = 16; m >= 1; m >>= 1) ss += __shfl_xor(ss, m, 32);

    float inv = 1.0f / fmaxf(sqrtf(ss), EPS);
    a.x *= inv; a.y *= inv; a.z *= inv; a.w *= inv;
    b.x *= inv; b.y *= inv; b.z *= inv; b.w *= inv;
    c.x *= inv; c.y *= inv; c.z *= inv; c.w *= inv;
    d.x *= inv; d.y *= inv; d.z *= inv; d.w *= inv;

    f32x4* o = (f32x4*)(out + (u64)t * FEAT);
    __builtin_nontemporal_store(a, &o[lane]);
    __builtin_nontemporal_store(b, &o[lane + 32]);
    __builtin_nontemporal_store(c, &o[lane + 64]);
    __builtin_nontemporal_store(d, &o[lane + 96]);
  }
}

// ---------------------------------------------------------------------------
// Fallback if workspace can't hold Wt (103 MB): direct strided column gather.
// ---------------------------------------------------------------------------
__global__ __launch_bounds__(256)
void gather_norm_direct_kernel(const int* __restrict__ ids,
                               const float* __restrict__ W,
                               const float* __restrict__ bias,
                               float* __restrict__ out, int n_tokens) {
  const int lane   = threadIdx.x & 31;
  const int wave   = (int)((blockIdx.x * blockDim.x + threadIdx.x) >> 5);
  const int nwaves = (int)((gridDim.x * blockDim.x) >> 5);

  for (int t = wave; t < n_tokens; t += nwaves) {
    int id = ids[t];
    float v[16];
    float ss = 0.f;
#pragma unroll
    for (int k = 0; k < 16; ++k) {
      int f = lane + 32 * k;
      float x = W[(u64)f * VOCAB + id] + bias[f];
      v[k] = x;
      ss += x * x;
    }
#pragma unroll
    for (int m = 16; m >= 1; m >>= 1) ss += __shfl_xor(ss, m, 32);
    float inv = 1.0f / fmaxf(sqrtf(ss), EPS);
#pragma unroll
    for (int k = 0; k < 16; ++k)
      __builtin_nontemporal_store(v[k] * inv, &out[(u64)t * FEAT + lane + 32 * k]);
  }
}

extern "C" void kernel_launch(void* const* d_in, const int* in_sizes, int n_in,
                              void* d_out, int out_size, void* d_ws, size_t ws_size,
                              hipStream_t stream) {
  const int*   ids  = (const int*)d_in[0];
  const float* W    = (const float*)d_in[1];
  const float* bias = (const float*)d_in[2];
  float*       out  = (float*)d_out;
  const int n_tokens = in_sizes[0];

  const size_t wt_bytes = (size_t)VOCAB * FEAT * sizeof(float);
  if (ws_size >= wt_bytes) {
    float* Wt = (float*)d_ws;
    dim3 tgrid((VOCAB + TDIM - 1) / TDIM, FEAT / TDIM);   // 786 x 8 tiles
    tdm_transpose_kernel<<<tgrid, 256, 0, stream>>>(W, Wt);
    gather_norm_kernel<<<4096, 256, 0, stream>>>(ids, Wt, bias, out, n_tokens);
  } else {
    gather_norm_direct_kernel<<<(n_tokens + 7) / 8, 256, 0, stream>>>(ids, W, bias, out, n_tokens);
  }
}